// DirectedEdgeDecoder_50345606644320
// MI455X (gfx1250) — compile-verified
//
#include <hip/hip_runtime.h>

typedef __attribute__((ext_vector_type(2))) float v2f;
typedef __attribute__((ext_vector_type(8))) float v8f;

#define DIM 128
#define HID 8
#define WROW 132   // LDS row stride (floats): 8B-aligned, bank-conflict-free

// ---------------------------------------------------------------------------
// Stage 1: per-node projection  P[n][h] = sum_k Z[n][k] * W[k][h]
// One wave computes a 16x16 output tile (only N=0..7 valid; B cols 8..15 = 0),
// accumulating K=128 via 32 V_WMMA_F32_16X16X4_F32 steps.
// W is staged once per block into LDS, transposed + zero-padded to 16 cols,
// so the inner loop is unpredicated: ds_load + global_load_b64 + wmma.
// blockIdx.y selects (z1, W1[0:128]) vs (z2, W1[128:256]).
// ---------------------------------------------------------------------------
__global__ void node_proj_wmma_kernel(const float* __restrict__ z1,
                                      const float* __restrict__ z2,
                                      const float* __restrict__ W1,
                                      float* __restrict__ P1,
                                      float* __restrict__ P2,
                                      int nNodes)
{
    const float* Z = (blockIdx.y == 0) ? z1 : z2;
    float*       P = (blockIdx.y == 0) ? P1 : P2;
    const float* W = W1 + (size_t)blockIdx.y * DIM * HID;  // W1 is [2*DIM, HID] row-major

    // ---- stage W into LDS: Wt[n][k] = W[k][n], zero for n >= HID ----
    __shared__ float Wt[16 * WROW];
    for (int i = threadIdx.x; i < 16 * DIM; i += blockDim.x) {
        const int n = i >> 7;            // / DIM
        const int k = i & (DIM - 1);
        Wt[n * WROW + k] = (n < HID) ? W[(size_t)k * HID + n] : 0.0f;
    }
    __syncthreads();

    const int wave   = threadIdx.x >> 5;
    const int lane   = threadIdx.x & 31;
    const int nTiles = (nNodes + 15) >> 4;
    const int tile   = blockIdx.x * (blockDim.x >> 5) + wave;
    if (tile >= nTiles) return;                 // wave-uniform exit (EXEC stays full)

    const int half  = lane >> 4;                // 0: lanes hold K=k..k+1 ; 1: K=k+2..k+3
    const int lrow  = lane & 15;                // A: M index ; B: N index
    const int mBase = tile * 16;

    int m = mBase + lrow;
    if (m >= nNodes) m = nNodes - 1;            // clamp padding rows (per-lane select)

    const float* zrow = Z  + (size_t)m * DIM + half * 2;
    const float* wrow = Wt + lrow * WROW + half * 2;

    v8f acc = {};
    #pragma unroll 4
    for (int k = 0; k < DIM; k += 4) {
        // A 16x4 f32: lanes 0-15 -> K=k,k+1 ; lanes 16-31 -> K=k+2,k+3
        v2f a = *(const v2f*)(zrow + k);        // global_load_b64 (8B aligned)
        // B 4x16 f32: same K striping, N across lanes; padded cols are zeros
        v2f b = *(const v2f*)(wrow + k);        // ds_load_b64 (8B aligned)
        acc = __builtin_amdgcn_wmma_f32_16x16x4_f32(
                  /*neg_a=*/false, a, /*neg_b=*/false, b,
                  /*c_mod=*/(short)0, acc, /*reuse_a=*/false, /*reuse_b=*/false);
    }

    // D layout: VGPR j, lanes 0-15 -> (M=mBase+j,    N=lane)
    //                   lanes 16-31 -> (M=mBase+8+j, N=lane-16)
    // Padding rows (mr >= nNodes) hold the projection of the clamped row
    // nNodes-1, so a clamped store writes the correct value idempotently:
    // stores are unconditional (no per-element exec masking).
    const int n = lrow;
    if (n < HID) {
        const int mOut = mBase + half * 8;
        #pragma unroll
        for (int j = 0; j < 8; j++) {
            int mr = mOut + j;
            if (mr > nNodes - 1) mr = nNodes - 1;   // v_min, not a branch
            P[(size_t)mr * HID + n] = acc[j];
        }
    }
}

// ---------------------------------------------------------------------------
// Stage 2: per-edge decode from L2-resident projection tables (64 B/edge).
// out[e] = relu(P1[row] + P2[col] + b1) . W2 + b2
// ---------------------------------------------------------------------------
__global__ void edge_decode_kernel(const long long* __restrict__ ei,
                                   const float* __restrict__ P1,
                                   const float* __restrict__ P2,
                                   const float* __restrict__ b1,
                                   const float* __restrict__ W2,
                                   const float* __restrict__ b2,
                                   float* __restrict__ out,
                                   int E)
{
    const int e = blockIdx.x * blockDim.x + threadIdx.x;
    if (e >= E) return;

    const long long r = ei[e];
    const long long c = ei[(size_t)E + e];

    const float4* p1 = (const float4*)(P1 + (size_t)r * HID);
    const float4* p2 = (const float4*)(P2 + (size_t)c * HID);
    float4 a0 = p1[0], a1 = p1[1];
    float4 d0 = p2[0], d1 = p2[1];

    float hv[8];
    hv[0] = a0.x + d0.x;  hv[1] = a0.y + d0.y;
    hv[2] = a0.z + d0.z;  hv[3] = a0.w + d0.w;
    hv[4] = a1.x + d1.x;  hv[5] = a1.y + d1.y;
    hv[6] = a1.z + d1.z;  hv[7] = a1.w + d1.w;

    float acc = b2[0];
    #pragma unroll
    for (int h = 0; h < HID; h++) {
        float v = fmaxf(hv[h] + b1[h], 0.0f);
        acc = fmaf(v, W2[h], acc);
    }
    out[e] = acc;
}

// ---------------------------------------------------------------------------
// Fallback (only if workspace is too small): fully fused per-edge decode.
// ---------------------------------------------------------------------------
__global__ void edge_decode_fused_kernel(const float* __restrict__ z1,
                                         const float* __restrict__ z2,
                                         const long long* __restrict__ ei,
                                         const float* __restrict__ W1,
                                         const float* __restrict__ b1,
                                         const float* __restrict__ W2,
                                         const float* __restrict__ b2,
                                         float* __restrict__ out,
                                         int E)
{
    const int e = blockIdx.x * blockDim.x + threadIdx.x;
    if (e >= E) return;
    const long long r = ei[e];
    const long long c = ei[(size_t)E + e];
    const float* zr = z1 + (size_t)r * DIM;
    const float* zc = z2 + (size_t)c * DIM;

    float h[HID];
    #pragma unroll
    for (int j = 0; j < HID; j++) h[j] = b1[j];
    for (int k = 0; k < DIM; k++) {
        const float v1 = zr[k], v2 = zc[k];
        #pragma unroll
        for (int j = 0; j < HID; j++) {
            h[j] = fmaf(v1, W1[(size_t)k * HID + j], h[j]);
            h[j] = fmaf(v2, W1[(size_t)(k + DIM) * HID + j], h[j]);
        }
    }
    float acc = b2[0];
    #pragma unroll
    for (int j = 0; j < HID; j++) acc = fmaf(fmaxf(h[j], 0.0f), W2[j], acc);
    out[e] = acc;
}

extern "C" void kernel_launch(void* const* d_in, const int* in_sizes, int n_in,
                              void* d_out, int out_size, void* d_ws, size_t ws_size,
                              hipStream_t stream)
{
    const float*     z1 = (const float*)d_in[0];
    const float*     z2 = (const float*)d_in[1];
    const long long* ei = (const long long*)d_in[2];   // int64 edge_index [2, E]
    const float*     W1 = (const float*)d_in[3];       // [2*DIM, HID]
    const float*     b1 = (const float*)d_in[4];       // [HID]
    const float*     W2 = (const float*)d_in[5];       // [HID, 1]
    const float*     b2 = (const float*)d_in[6];       // [1]
    float*          out = (float*)d_out;

    const int nNodes = in_sizes[0] / DIM;              // 100000
    const int E      = in_sizes[2] / 2;                // 800000

    const size_t needWs = 2 * (size_t)nNodes * HID * sizeof(float);  // 6.4 MB

    if (ws_size >= needWs) {
        float* P1 = (float*)d_ws;
        float* P2 = P1 + (size_t)nNodes * HID;

        const int nTiles        = (nNodes + 15) / 16;
        const int wavesPerBlock = 8;                   // 256 threads = 8 wave32
        dim3 grid((nTiles + wavesPerBlock - 1) / wavesPerBlock, 2);
        node_proj_wmma_kernel<<<grid, 256, 0, stream>>>(z1, z2, W1, P1, P2, nNodes);

        edge_decode_kernel<<<(E + 255) / 256, 256, 0, stream>>>(
            ei, P1, P2, b1, W2, b2, out, E);
    } else {
        edge_decode_fused_kernel<<<(E + 255) / 256, 256, 0, stream>>>(
            z1, z2, ei, W1, b1, W2, b2, out, E);
    }
}